// RandomBitFlip_6442450944425
// MI455X (gfx1250) — compile-verified
//
#include <hip/hip_runtime.h>
#include <stdint.h>

// 4 x u32 native vector => global_load_b128 / ds_load_b128 / global_store_b128
typedef uint32_t v4u __attribute__((ext_vector_type(4)));

#define BF_THREADS 256
#define BF_DEPTH   4          // async LDS ring depth (4 x 4KB = 16KB / block)
#define BF_BUFBYTES 4096u     // BF_THREADS * 16B per buffer

// ---------------------------------------------------------------------------
// Bernoulli(p = 2^-14) per-bit flips over a 128-bit chunk, sampled sparsely.
// The reference's exponent mask (0xFF800000) and fraction mask (0x007FFFFF)
// partition all 32 bits, so one Bernoulli-per-bit word over the union is
// distribution-equivalent to the two separate streams.
// Cumulative Binomial(128, 2^-14) thresholds scaled to 2^32:
//   P(0 flips)  = 0.9922177  -> T0
//   P(<=1 flip) = 0.9999699  -> T1
//   P(<=2)      = 0.99999992 -> T2   (P(>=4) ~ 1.5e-10, ignored)
// ---------------------------------------------------------------------------
#define BF_T0 4261542579u
#define BF_T1 4294837914u
#define BF_T2 4294966965u

__device__ __forceinline__ uint32_t lowbias32(uint32_t x) {
  x ^= x >> 16; x *= 0x7feb352du;
  x ^= x >> 15; x *= 0x846ca68bu;
  x ^= x >> 16;
  return x;
}

// Set bit `pos` (0..127) inside the 4-word mask, branchlessly.
__device__ __forceinline__ void add_flip(uint32_t &m0, uint32_t &m1,
                                         uint32_t &m2, uint32_t &m3,
                                         uint32_t pos) {
  uint32_t one = 1u << (pos & 31u);
  uint32_t w   = (pos >> 5) & 3u;
  m0 |= (w == 0u) ? one : 0u;
  m1 |= (w == 1u) ? one : 0u;
  m2 |= (w == 2u) ? one : 0u;
  m3 |= (w == 3u) ? one : 0u;
}

__device__ __forceinline__ void chunk_masks(uint32_t chunkId,
                                            uint32_t &m0, uint32_t &m1,
                                            uint32_t &m2, uint32_t &m3) {
  uint32_t u = lowbias32(chunkId ^ 0xA511E9B3u);
  if (u >= BF_T0) {                            // >=1 flip: ~0.78% of chunks
    uint32_t r = lowbias32(u ^ 0x9E3779B9u);   // fresh bits for positions
    add_flip(m0, m1, m2, m3, r & 127u);
    if (u >= BF_T1) add_flip(m0, m1, m2, m3, (r >> 7) & 127u);
    if (u >= BF_T2) add_flip(m0, m1, m2, m3, (r >> 14) & 127u);
  }
}

// zero_out_invalid: exponent field all-ones (inf/nan) -> +0.0f
__device__ __forceinline__ uint32_t zero_nonfinite(uint32_t x) {
  return ((x & 0x7F800000u) == 0x7F800000u) ? 0u : x;
}

__device__ __forceinline__ v4u process_chunk(v4u v, uint32_t chunkId) {
  uint32_t m0 = 0u, m1 = 0u, m2 = 0u, m3 = 0u;
  chunk_masks(chunkId, m0, m1, m2, m3);
  v.x = zero_nonfinite(v.x ^ m0);
  v.y = zero_nonfinite(v.y ^ m1);
  v.z = zero_nonfinite(v.z ^ m2);
  v.w = zero_nonfinite(v.w ^ m3);
  return v;
}

// ---------------------------------------------------------------------------
// CDNA5 async data movement: per-lane 16B global -> LDS, tracked by ASYNCcnt.
// Input stream is read exactly once -> NT so it does not displace the 192MB L2.
// ---------------------------------------------------------------------------
__device__ __forceinline__ void async_load_b128_to_lds(uint32_t ldsByteAddr,
                                                       const void* gaddr) {
  asm volatile("global_load_async_to_lds_b128 %0, %1, off th:TH_LOAD_NT"
               :: "v"(ldsByteAddr), "v"(gaddr)
               : "memory");
}

// Main streaming kernel. Requires uniform iters >= BF_DEPTH (launcher enforces):
// EXEC stays all-ones, waits are constant immediates. 4-deep LDS ring so each
// wave keeps 4 x 512B async transfers in flight (~3 iterations of prefetch
// distance on top of multi-wave occupancy to cover HBM latency).
__global__ __launch_bounds__(BF_THREADS) void
bitflip_async(const uint32_t* __restrict__ in, uint32_t* __restrict__ out,
              uint32_t iters) {
  __shared__ __attribute__((aligned(16))) uint32_t ldsbuf[BF_DEPTH][BF_THREADS * 4];
  const uint32_t tid    = threadIdx.x;
  const uint32_t stride = gridDim.x * blockDim.x;       // chunks per sweep
  uint32_t chunk = blockIdx.x * blockDim.x + tid;

  const uint32_t* gsrc = in  + (size_t)chunk * 4u;
  uint32_t*       gdst = out + (size_t)chunk * 4u;
  const size_t   gstep = (size_t)stride * 4u;

  const uint32_t ldsBase = (uint32_t)(uintptr_t)&ldsbuf[0][tid * 4u];

  // Prologue: fill buffers 0..2 (valid since iters >= BF_DEPTH).
  async_load_b128_to_lds(ldsBase + 0u * BF_BUFBYTES, gsrc); gsrc += gstep;
  async_load_b128_to_lds(ldsBase + 1u * BF_BUFBYTES, gsrc); gsrc += gstep;
  async_load_b128_to_lds(ldsBase + 2u * BF_BUFBYTES, gsrc); gsrc += gstep;

  for (uint32_t k = 0; k < iters; ++k) {
    if (k + (BF_DEPTH - 1u) < iters) {
      // Issue chunk k+3 into the ring slot that was consumed at iteration k-1,
      // then wait until at most the 3 younger transfers remain outstanding
      // (i.e. the current buffer's load has completed).
      async_load_b128_to_lds(ldsBase + (((k + 3u) & 3u) << 12), gsrc);
      gsrc += gstep;
      asm volatile("s_wait_asynccnt 3" ::: "memory");
    } else {
      // Drain: all remaining outstanding loads target future buffers anyway.
      asm volatile("s_wait_asynccnt 0" ::: "memory");
    }
    v4u v = *reinterpret_cast<const v4u*>(&ldsbuf[k & 3u][tid * 4u]); // ds_load_b128
    v = process_chunk(v, chunk);
    __builtin_nontemporal_store(v, reinterpret_cast<v4u*>(gdst));     // b128 th:NT
    gdst  += gstep;
    chunk += stride;
  }
}

// Remainder path: direct NT b128 loads over chunks [chunkBase, chunkEnd) plus
// a scalar tail for n % 4 elements. Also exercises global_prefetch_b8.
__global__ __launch_bounds__(BF_THREADS) void
bitflip_direct(const uint32_t* __restrict__ in, uint32_t* __restrict__ out,
               uint32_t chunkBase, uint32_t chunkEnd, uint32_t n) {
  const uint32_t gid    = blockIdx.x * blockDim.x + threadIdx.x;
  const uint32_t stride = gridDim.x * blockDim.x;

  for (uint32_t c = chunkBase + gid; c < chunkEnd; c += stride) {
    const v4u* p = reinterpret_cast<const v4u*>(in + (size_t)c * 4u);
    __builtin_prefetch(reinterpret_cast<const v4u*>(in + ((size_t)c + stride) * 4u), 0, 0);
    v4u v = __builtin_nontemporal_load(p);
    v = process_chunk(v, c);
    __builtin_nontemporal_store(v, reinterpret_cast<v4u*>(out + (size_t)c * 4u));
  }

  // Scalar tail: elements [chunkEnd*4, n)
  const uint32_t base = chunkEnd << 2;
  for (uint32_t e = base + gid; e < n; e += stride) {
    uint32_t c = e >> 2, w = e & 3u;
    uint32_t m0 = 0u, m1 = 0u, m2 = 0u, m3 = 0u;
    chunk_masks(c, m0, m1, m2, m3);
    uint32_t mm = (w == 0u) ? m0 : ((w == 1u) ? m1 : ((w == 2u) ? m2 : m3));
    out[e] = zero_nonfinite(in[e] ^ mm);
  }
}

extern "C" void kernel_launch(void* const* d_in, const int* in_sizes, int n_in,
                              void* d_out, int out_size, void* d_ws, size_t ws_size,
                              hipStream_t stream) {
  (void)n_in; (void)out_size; (void)d_ws; (void)ws_size;
  const uint32_t n  = (uint32_t)in_sizes[0];     // fp32 element count
  const uint32_t n4 = n >> 2;                    // 128-bit chunks
  const uint32_t* in  = (const uint32_t*)d_in[0];
  uint32_t*       out = (uint32_t*)d_out;

  const int threads = BF_THREADS;
  const int blocks  = 4096;                      // 32K wave32 waves in flight
  const uint32_t stride = (uint32_t)blocks * (uint32_t)threads;  // 2^20 chunks

  uint32_t iters = n4 / stride;                  // uniform per-lane trip count
  if (iters < BF_DEPTH) iters = 0;               // async kernel needs >= DEPTH
  const uint32_t mainChunks = iters * stride;
  if (iters > 0) {
    bitflip_async<<<blocks, threads, 0, stream>>>(in, out, iters);
  }

  const uint32_t remChunks = n4 - mainChunks;
  const uint32_t tailElems = n - (n4 << 2);
  if (remChunks > 0u || tailElems > 0u) {
    uint32_t work = remChunks + tailElems;
    int b2 = (int)((work + (uint32_t)threads - 1u) / (uint32_t)threads);
    if (b2 < 1) b2 = 1;
    bitflip_direct<<<b2, threads, 0, stream>>>(in, out, mainChunks, n4, n);
  }
}